// PowerSpectrum_6279242187188
// MI455X (gfx1250) — compile-verified
//
#include <hip/hip_runtime.h>

// Wave32 WMMA fragments for V_WMMA_F32_16X16X4_F32:
//   A: 16x4 f32 -> 2 VGPRs/lane ; B: 4x16 f32 -> 2 VGPRs/lane ; C/D: 16x16 f32 -> 8 VGPRs/lane
typedef __attribute__((ext_vector_type(2))) float v2f;
typedef __attribute__((ext_vector_type(8))) float v8f;

// (2l+1)^{-1/2}, indexed by compile-time L -> folds to an inline literal.
__device__ constexpr float CG_TAB[6] = {
    1.0f, 0.57735026918962576f, 0.44721359549995794f,
    0.37796447300922723f, 0.33333333333333333f, 0.30151134457776363f};

// out[n, L*4096 + f*64 + g] = (2L+1)^{-1/2} * sum_m c[n,m,f] * c[n,m,g]
// One wave computes one 16x16 tile (f0,g0) of the 64x64 Gram matrix of one sample n.
template <int L>
__global__ __launch_bounds__(256) void ps_wmma_l(const float* __restrict__ c,
                                                 float* __restrict__ out,
                                                 int nSamples)
{
    constexpr int M     = 2 * L + 1;        // K extent (1,3,5,7,9,11)
    constexpr int KFULL = (M / 4) * 4;      // unmasked 4-wide K steps
    constexpr int TAIL0 = KFULL;            // tail base row (remainder is 1 or 3 rows)

    const int lane    = threadIdx.x & 31;
    const long long w = (long long)blockIdx.x * 8 + (threadIdx.x >> 5); // wave id
    const int tile    = (int)(w & 15);      // 4x4 grid of 16x16 tiles
    const long long n = w >> 4;
    if (n >= nSamples) return;              // wave-uniform; active waves keep EXEC all-1s

    const float* __restrict__ cb = c + n * (long long)(M * 64);  // c[n,:,:]

    const int tf = tile >> 2, tg = tile & 3;
    const int f0 = tf << 4,  g0 = tg << 4;
    const int t    = lane & 15;             // fragment column
    const int half = lane >> 4;             // K-pair select: lanes 0-15 hold K={m0,m0+1}, 16-31 {m0+2,m0+3}

    v8f acc = {};

    // Full K-steps: all rows in range, straight-line unconditional loads, no selects.
#pragma unroll
    for (int m0 = 0; m0 < KFULL; m0 += 4) {
        const int ma = m0 + 2 * half;
        v2f a, b;
        a.x = cb[(ma    ) * 64 + f0 + t];
        a.y = cb[(ma + 1) * 64 + f0 + t];
        b.x = cb[(ma    ) * 64 + g0 + t];
        b.y = cb[(ma + 1) * 64 + g0 + t];
        acc = __builtin_amdgcn_wmma_f32_16x16x4_f32(
                  false, a, false, b, (short)0, acc, false, false);
    }

    // Tail step (K % 4 == 1 or 3): unconditional loads from clamped row addresses,
    // then per-lane zero-select (v_cndmask) -- no divergent branches, EXEC untouched.
    {
        const int r0  = TAIL0 + 2 * half;
        const int r1  = r0 + 1;
        const int r0c = (r0 < M) ? r0 : (M - 1);
        const int r1c = (r1 < M) ? r1 : (M - 1);
        const float ax = cb[r0c * 64 + f0 + t];
        const float ay = cb[r1c * 64 + f0 + t];
        const float bx = cb[r0c * 64 + g0 + t];
        const float by = cb[r1c * 64 + g0 + t];
        v2f a, b;
        a.x = (r0 < M) ? ax : 0.0f;
        a.y = (r1 < M) ? ay : 0.0f;
        b.x = (r0 < M) ? bx : 0.0f;
        b.y = (r1 < M) ? by : 0.0f;
        acc = __builtin_amdgcn_wmma_f32_16x16x4_f32(
                  false, a, false, b, (short)0, acc, false, false);
    }

    constexpr float cg = CG_TAB[L];

    // C/D layout: VGPR j -> row (j + 8*half), lane t -> column.
    // Non-temporal streaming stores: 983 MB write-once output must not thrash L2
    // (inputs, 92 MB, stay L2-resident for the 8x tile reuse).
    float* __restrict__ ob = out + n * 24576LL + L * 4096 + f0 * 64 + g0;
#pragma unroll
    for (int j = 0; j < 8; ++j) {
        __builtin_nontemporal_store(cg * acc[j], &ob[(j + 8 * half) * 64 + t]);
    }
}

extern "C" void kernel_launch(void* const* d_in, const int* in_sizes, int n_in,
                              void* d_out, int out_size, void* d_ws, size_t ws_size,
                              hipStream_t stream)
{
    (void)n_in; (void)d_ws; (void)ws_size; (void)out_size;
    float* out = (float*)d_out;
    const int N = in_sizes[0] / 64;            // c0 is (N, 1, 64)

    // One wave per 16x16 tile: N*16 waves per l -> 2N blocks of 256 threads (8 waves).
    const int blocks = (N * 16 + 7) / 8;
    ps_wmma_l<0><<<blocks, 256, 0, stream>>>((const float*)d_in[0], out, N);
    ps_wmma_l<1><<<blocks, 256, 0, stream>>>((const float*)d_in[1], out, N);
    ps_wmma_l<2><<<blocks, 256, 0, stream>>>((const float*)d_in[2], out, N);
    ps_wmma_l<3><<<blocks, 256, 0, stream>>>((const float*)d_in[3], out, N);
    ps_wmma_l<4><<<blocks, 256, 0, stream>>>((const float*)d_in[4], out, N);
    ps_wmma_l<5><<<blocks, 256, 0, stream>>>((const float*)d_in[5], out, N);
}